// YoloLoss_42271068127491
// MI455X (gfx1250) — compile-verified
//
#include <hip/hip_runtime.h>
#include <math.h>

// ---------------- problem constants ----------------
#define BATCH 32
#define NT    32          // targets per image
#define S     52
#define SS    (S*S)       // 2704
#define NCELLS (BATCH*3*SS)   // 259584 == 1014 * 256 exactly
#define NCLS  80
#define SUBA  6
#define INV_B (1.0f/32.0f)

typedef float v2f __attribute__((ext_vector_type(2)));
typedef float v8f __attribute__((ext_vector_type(8)));

// anchors / stride (stride = 416/52 = 8)
__constant__ float AW9[9] = {14.5f, 19.5f, 46.625f, 3.75f, 7.75f, 7.375f, 1.25f, 2.0f, 4.125f};
__constant__ float AH9[9] = {11.25f, 24.75f, 40.75f, 7.625f, 5.625f, 14.875f, 1.625f, 3.75f, 2.875f};

__device__ __forceinline__ float sigmoidf_(float x) {
    return 1.0f / (1.0f + __expf(-x));
}
__device__ __forceinline__ float bcef_(float p, float t) {
    p = fminf(fmaxf(p, 1e-7f), 1.0f - 1e-7f);
    return -t * __logf(p) - (1.0f - t) * __logf(1.0f - p);
}

// ---------------- kernel 0: init ----------------
__global__ void k_init(int* __restrict__ cellTarget, float* __restrict__ res) {
    int idx = blockIdx.x * 256 + threadIdx.x;
    if (idx < NCELLS) cellTarget[idx] = -1;
    if (idx < 2) res[idx] = 0.0f;
}

// ---------------- kernel 1: per-target prep + scatter ----------------
// targets: (B, NT, 5) = (x, y, w, h, cls) in image units
__global__ void k_prep(const float* __restrict__ targets,
                       int*   __restrict__ cellTarget,
                       float4* __restrict__ iouBox,
                       float* __restrict__ iouArea,
                       float* __restrict__ matchRec) {
    int t = blockIdx.x * 64 + threadIdx.x;     // 16 blocks * 64 = 1024
    if (t >= BATCH * NT) return;
    const float* tp = targets + t * 5;
    float txi = tp[0], tyi = tp[1], twi = tp[2], thi = tp[3], cls = tp[4];
    float gx = txi * 0.125f, gy = tyi * 0.125f;   // /stride
    float gw = twi * 0.125f, gh = thi * 0.125f;

    // box (grid units) for per-cell IoU ignore test — used for ALL targets
    iouBox[t]  = make_float4(gx - 0.5f * gw, gy - 0.5f * gh,
                             gx + 0.5f * gw, gy + 0.5f * gh);
    iouArea[t] = gw * gh;

    // best anchor (first-max tie break like jnp.argmax)
    int best = 0; float br = -1.0f;
#pragma unroll
    for (int a = 0; a < 9; ++a) {
        float aw = AW9[a], ah = AH9[a];
        float inter = fminf(gw, aw) * fminf(gh, ah);
        float uni   = gw * gh + aw * ah - inter;
        float r = inter / uni;
        if (r > br) { br = r; best = a; }
    }
    int gi = (int)floorf(gx);
    int gj = (int)floorf(gy);

    float* m = matchRec + t * 8;
    m[0] = gx - (float)gi;                         // tx
    m[1] = gy - (float)gj;                         // ty
    m[2] = __logf(gw / AW9[best]);                 // tw
    m[3] = __logf(gh / AH9[best]);                 // th
    m[4] = 2.0f - (twi / 416.0f) * (thi / 416.0f); // box_loss_scale
    m[5] = cls;
    m[6] = 0.0f; m[7] = 0.0f;

    if (best >= SUBA && best < SUBA + 3) {         // valid
        int b = t >> 5;                            // t / NT
        int k = best - SUBA;
        int flat = ((b * 3 + k) * S + gj) * S + gi;
        cellTarget[flat] = t;                      // non-atomic: last-write-wins, matches .at[].set
    }
}

// ---------------- kernel 2: main streaming loss ----------------
__global__ __launch_bounds__(256)
void k_main(const float* __restrict__ out,
            const int*   __restrict__ cellTarget,
            const float4* __restrict__ iouBox,
            const float* __restrict__ iouArea,
            const float* __restrict__ matchRec,
            float* __restrict__ res) {
    int cell = blockIdx.x * 256 + threadIdx.x;     // grid exactly covers NCELLS
    int i =  cell % S;
    int j = (cell / S) % S;
    int a = (cell / SS) % 3;
    int b =  cell / (3 * SS);

    // pred[b,a,j,i,c] = out[b, a*85+c, j, i]
    int base = ((b * 255 + a * 85) * SS) + j * S + i;
    float xl = out[base];
    float yl = out[base + SS];
    float wl = out[base + 2 * SS];
    float hl = out[base + 3 * SS];
    float clg = out[base + 4 * SS];

    float sx = sigmoidf_(xl);
    float sy = sigmoidf_(yl);
    float sconf = sigmoidf_(clg);

    float law = (a == 0) ? 1.25f  : (a == 1) ? 2.0f  : 4.125f;
    float lah = (a == 0) ? 1.625f : (a == 1) ? 3.75f : 2.875f;
    float pw = __expf(wl) * law;
    float ph = __expf(hl) * lah;
    float px = sx + (float)i;
    float py = sy + (float)j;
    float bx1 = px - 0.5f * pw, bx2 = px + 0.5f * pw;
    float by1 = py - 0.5f * ph, by2 = py + 0.5f * ph;
    float areab = pw * ph;

    // max IoU vs all 32 targets of this batch (targets are wave-uniform -> SGPR loads)
    float ioumax = 0.0f;
    const float4* tb = iouBox  + b * NT;
    const float*  ta = iouArea + b * NT;
#pragma unroll 8
    for (int n = 0; n < NT; ++n) {
        float4 t4 = tb[n];
        float iw = fminf(t4.z, bx2) - fmaxf(t4.x, bx1);
        float ih = fminf(t4.w, by2) - fmaxf(t4.y, by1);
        iw = fmaxf(iw, 0.0f); ih = fmaxf(ih, 0.0f);
        float inter = iw * ih;
        float iou = inter / (ta[n] + areab - inter);
        ioumax = fmaxf(ioumax, iou);
    }

    int rec = cellTarget[cell];
    float maskv = (rec >= 0) ? 1.0f : 0.0f;
    float noobj = (rec >= 0) ? 0.0f : ((ioumax > 0.5f) ? 0.0f : 1.0f);

    float bconf = bcef_(sconf, maskv);
    float loss = bconf * (maskv + noobj) * INV_B;
    float cnt = maskv;

    if (rec >= 0) {
        const float* m = matchRec + rec * 8;
        float tx = m[0], ty = m[1], tw = m[2], th = m[3], scale = m[4];
        int cid = (int)m[5];
        loss += (bcef_(sx, tx) + bcef_(sy, ty)) * scale * INV_B;
        float dw = wl - tw, dh = hl - th;
        loss += 0.5f * (dw * dw + dh * dh) * scale * INV_B;
        const float* cp = out + base + 5 * SS;
        // pull the 80 strided class logits into cache early (-> global_prefetch_b8)
#pragma unroll
        for (int c = 0; c < NCLS; c += 8) __builtin_prefetch(cp + c * SS, 0, 0);
#pragma unroll 4
        for (int c = 0; c < NCLS; ++c) {
            float p = sigmoidf_(cp[c * SS]);
            float tt = (c == cid) ? 1.0f : 0.0f;
            loss += bcef_(p, tt) * INV_B;
        }
    }

    // ---- wave reduction of (loss, cnt) with V_WMMA_F32_16X16X4_F32 ----
    // A (16x4 f32): lane L supplies rows M=L%16; {K0,K1} lanes 0-15, {K2,K3} lanes 16-31.
    // Put (loss, cnt) per lane; selector B (rows K0,K2 vs K1,K3) reduces each separately.
    int lane = threadIdx.x & 31;
    float wavL, wavC;
#if __has_builtin(__builtin_amdgcn_wmma_f32_16x16x4_f32)
    {
        v2f av; av[0] = loss; av[1] = cnt;
        float sel = (lane < 16) ? 1.0f : 0.0f;     // B vgpr0: row0|row1 ; vgpr1: row2|row3
        v2f bl; bl[0] = sel;        bl[1] = sel;          // picks K0,K2 -> loss
        v2f bc; bc[0] = 1.0f - sel; bc[1] = 1.0f - sel;   // picks K1,K3 -> cnt
        v8f cz = {};
        v8f dl = __builtin_amdgcn_wmma_f32_16x16x4_f32(false, av, false, bl,
                                                       (short)0, cz, false, false);
        v8f dc = __builtin_amdgcn_wmma_f32_16x16x4_f32(false, av, false, bc,
                                                       (short)0, cz, false, false);
        // D: vgpr r = row M=r (lanes 0-15) / M=r+8 (lanes 16-31); columns identical
        float sL = dl[0]+dl[1]+dl[2]+dl[3]+dl[4]+dl[5]+dl[6]+dl[7];
        float sC = dc[0]+dc[1]+dc[2]+dc[3]+dc[4]+dc[5]+dc[6]+dc[7];
        wavL = __shfl(sL, 0) + __shfl(sL, 16);
        wavC = __shfl(sC, 0) + __shfl(sC, 16);
    }
#else
    {
        float sL = loss, sC = cnt;
        for (int d = 16; d > 0; d >>= 1) {
            sL += __shfl_xor(sL, d);
            sC += __shfl_xor(sC, d);
        }
        wavL = sL; wavC = sC;
    }
#endif

    __shared__ float redL[8];
    __shared__ float redC[8];
    int wid = threadIdx.x >> 5;
    if (lane == 0) { redL[wid] = wavL; redC[wid] = wavC; }
    __syncthreads();
    if (threadIdx.x == 0) {
        float L = 0.0f, C = 0.0f;
#pragma unroll
        for (int w = 0; w < 8; ++w) { L += redL[w]; C += redC[w]; }
        atomicAdd(&res[0], L);
        atomicAdd(&res[1], C);
    }
}

// ---------------- kernel 3: finalize ----------------
__global__ void k_fin(float* __restrict__ res) {
    if (threadIdx.x == 0) res[1] = fmaxf(res[1], 1.0f);
}

// ---------------- host launcher ----------------
extern "C" void kernel_launch(void* const* d_in, const int* in_sizes, int n_in,
                              void* d_out, int out_size, void* d_ws, size_t ws_size,
                              hipStream_t stream) {
    const float* out_t   = (const float*)d_in[0];   // (32,255,52,52)
    const float* targets = (const float*)d_in[1];   // (32,32,5)
    float* res = (float*)d_out;                     // [loss, num_pos]

    char* ws = (char*)d_ws;
    size_t off = 0;
    int* cellTarget = (int*)(ws + off);            off += (size_t)NCELLS * sizeof(int);      // 1,038,336 B
    float4* iouBox  = (float4*)(ws + off);         off += (size_t)BATCH * NT * sizeof(float4);
    float* iouArea  = (float*)(ws + off);          off += (size_t)BATCH * NT * sizeof(float);
    // pad to 16B
    off = (off + 15) & ~(size_t)15;
    float* matchRec = (float*)(ws + off);          off += (size_t)BATCH * NT * 8 * sizeof(float);
    (void)off; (void)ws_size; (void)in_sizes; (void)n_in; (void)out_size;

    k_init<<<(NCELLS + 255) / 256, 256, 0, stream>>>(cellTarget, res);
    k_prep<<<(BATCH * NT + 63) / 64, 64, 0, stream>>>(targets, cellTarget, iouBox, iouArea, matchRec);
    k_main<<<NCELLS / 256, 256, 0, stream>>>(out_t, cellTarget, iouBox, iouArea, matchRec, res);
    k_fin<<<1, 32, 0, stream>>>(res);
}